// MMFeature_27754078667621
// MI455X (gfx1250) — compile-verified
//
#include <hip/hip_runtime.h>
#include <hip/hip_bf16.h>
#include <math.h>

// ---------------------------------------------------------------------------
// MMFeature attention aggregation for MI455X (gfx1250, wave32).
//   o      = att_rel @ u_w + u_b                (WMMA f32 16x16x4 matvec)
//   alpha  = segment_softmax(o, att_ids)        (ids sorted -> deterministic)
//   text   = segment_sum(alpha * [att_rel|att]) (one block per entity)
//   out    = [text (N,256) , img_features (N,2048)]  (D2D async copy)
// Bandwidth-bound: ~2.5 GB traffic/call -> ~107 us floor at 23.3 TB/s.
// ---------------------------------------------------------------------------

typedef float v2f __attribute__((ext_vector_type(2)));
typedef float v8f __attribute__((ext_vector_type(8)));

// ---------------------------------------------------------------------------
// Kernel 1: o[e] = dot(att_rel[e,:], u_w) + u_b via V_WMMA_F32_16X16X4_F32.
// One wave handles 16 rows. A fragment (16x4 f32): lane%16 = M, lanes>=16
// supply K=2,3 (ISA 7.12.2 32-bit A layout). B fragment broadcasts the same
// u_w chunk into every column, so D[m, n] == dot(row m) for all n; lane 0
// holds rows 0..7 in c[0..7], lane 16 holds rows 8..15.
// ---------------------------------------------------------------------------
__global__ void __launch_bounds__(128)
o_matvec_wmma(const float* __restrict__ att_rel,
              const float* __restrict__ u_w,
              const float* __restrict__ u_b,
              float* __restrict__ o, int E)
{
    const int lane    = threadIdx.x & 31;
    const int wave    = threadIdx.x >> 5;
    const int rowBase = (blockIdx.x * 4 + wave) * 16;
    if (rowBase >= E) return;                     // wave-uniform: EXEC stays full

    if (rowBase + 16 <= E) {
        // -------- WMMA path (full 16-row tile) --------
        const int m     = lane & 15;              // row within tile
        const int khalf = (lane & 16) >> 3;       // lanes 0-15 -> k+0,1 ; 16-31 -> k+2,3
        const float* rowp = att_rel + (size_t)(rowBase + m) * 128;

        v8f c = {0.f, 0.f, 0.f, 0.f, 0.f, 0.f, 0.f, 0.f};
        #pragma unroll 8
        for (int kk = 0; kk < 32; ++kk) {
            const int k = (kk << 2) + khalf;      // even -> 8B aligned
            v2f a = *(const v2f*)(rowp + k);
            v2f b = *(const v2f*)(u_w + k);       // same value in every column
            c = __builtin_amdgcn_wmma_f32_16x16x4_f32(
                    /*neg_a=*/false, a, /*neg_b=*/false, b,
                    /*c_mod=*/(short)0, c, /*reuse_a=*/false, /*reuse_b=*/false);
        }
        const float bias = u_b[0];
        if (lane == 0) {
            #pragma unroll
            for (int r = 0; r < 8; ++r) o[rowBase + r] = c[r] + bias;
        } else if (lane == 16) {
            #pragma unroll
            for (int r = 0; r < 8; ++r) o[rowBase + 8 + r] = c[r] + bias;
        }
    } else {
        // -------- scalar tail (E % 16 rows) --------
        const int r = rowBase + lane;
        if (r < E) {
            const float* rowp = att_rel + (size_t)r * 128;
            float acc = 0.f;
            #pragma unroll 4
            for (int k = 0; k < 128; ++k) acc += rowp[k] * u_w[k];
            o[r] = acc + u_b[0];
        }
    }
}

// ---------------------------------------------------------------------------
// Kernel 2: per-entity segment softmax + weighted feature sum.
// att_ids is sorted ascending, so each entity's attributes are contiguous;
// binary search gives [lo, hi). Fixed summation order -> deterministic.
// blockDim = 256: thread d produces text[n, d] (d<128 -> att_rel, else att).
// ---------------------------------------------------------------------------
__device__ __forceinline__ int lower_bound_i32(const int* __restrict__ a,
                                               int n, int key)
{
    int lo = 0, hi = n;
    while (lo < hi) {
        int mid = (lo + hi) >> 1;
        if (a[mid] < key) lo = mid + 1; else hi = mid;
    }
    return lo;
}

__global__ void __launch_bounds__(256)
seg_softmax_scatter(const float* __restrict__ o,
                    const float* __restrict__ att_rel,
                    const float* __restrict__ att,
                    const int*   __restrict__ ids,
                    float* __restrict__ text,
                    int n_att)
{
    const int n = blockIdx.x;

    __shared__ int   s_lo, s_len;
    __shared__ float s_max, s_denom;
    __shared__ float s_w[32];

    if (threadIdx.x == 0) {
        const int lo = lower_bound_i32(ids, n_att, n);
        const int hi = lo + lower_bound_i32(ids + lo, n_att - lo, n + 1);
        s_lo  = lo;
        s_len = hi - lo;
        float m = -INFINITY;
        for (int j = lo; j < hi; ++j) m = fmaxf(m, o[j]);
        float dsum = 0.f;
        for (int j = lo; j < hi; ++j) dsum += expf(o[j] - m);
        s_max   = m;
        s_denom = dsum;
    }
    __syncthreads();

    const int   lo  = s_lo;
    const int   L   = s_len;
    const float mx  = s_max;
    const float inv = (L > 0) ? (1.0f / s_denom) : 0.f;

    const int d  = threadIdx.x;
    const float* fbase = (d < 128) ? att_rel : att;   // concat order: [rel | att]
    const int   dd = (d < 128) ? d : (d - 128);

    float acc = 0.f;
    for (int c0 = 0; c0 < L; c0 += 32) {
        const int cl = min(32, L - c0);
        __syncthreads();                               // protect s_w reuse
        if (threadIdx.x < cl)
            s_w[threadIdx.x] = expf(o[lo + c0 + threadIdx.x] - mx) * inv;
        __syncthreads();
        for (int j = 0; j < cl; ++j)
            acc += s_w[j] * fbase[(size_t)(lo + c0 + j) * 128 + dd];
    }
    text[(size_t)n * 256 + d] = acc;
}

// ---------------------------------------------------------------------------
// Launch. Inputs (setup_inputs order):
//   0 img_features [N,2048] f32   1 att_features [E,128] f32
//   2 att_rel_features [E,128]    3 att_ids [E] int32 (x64 off)
//   4 u_w [128,1] f32             5 u_b [1] f32
// Output: text (N*256 f32) ++ img_features (N*2048 f32).
// ---------------------------------------------------------------------------
extern "C" void kernel_launch(void* const* d_in, const int* in_sizes, int n_in,
                              void* d_out, int out_size, void* d_ws, size_t ws_size,
                              hipStream_t stream)
{
    const float* img     = (const float*)d_in[0];
    const float* att     = (const float*)d_in[1];
    const float* att_rel = (const float*)d_in[2];
    const int*   ids     = (const int*)d_in[3];
    const float* u_w     = (const float*)d_in[4];
    const float* u_b     = (const float*)d_in[5];

    const int    n_att     = in_sizes[3];                 // 500000
    const int    tdim      = in_sizes[4];                 // 128
    const size_t img_elems = (size_t)in_sizes[0];         // N * 2048
    const int    n_ent     = (int)(((size_t)out_size - img_elems) / (2 * tdim));

    float* o    = (float*)d_ws;                           // n_att floats scratch
    float* text = (float*)d_out;                          // [n_ent, 256]
    float* imgo = (float*)d_out + (size_t)n_ent * (2 * tdim);

    // 1) o = att_rel @ u_w + u_b   (WMMA f32)
    {
        const int rowsPerBlock = 64;                      // 4 waves x 16 rows
        const int grid = (n_att + rowsPerBlock - 1) / rowsPerBlock;
        o_matvec_wmma<<<grid, 128, 0, stream>>>(att_rel, u_w, u_b, o, n_att);
    }

    // 2) segment softmax + weighted scatter-sum (one block per entity)
    seg_softmax_scatter<<<n_ent, 256, 0, stream>>>(o, att_rel, att, ids, text, n_att);

    // 3) img_features passthrough (819 MB D2D, graph-capture safe)
    hipMemcpyAsync(imgo, img, img_elems * sizeof(float),
                   hipMemcpyDeviceToDevice, stream);
}